// qlayer1_91018946937207
// MI455X (gfx1250) — compile-verified
//
#include <hip/hip_runtime.h>
#include <hip/hip_bf16.h>

typedef __attribute__((ext_vector_type(2))) float v2f;
typedef __attribute__((ext_vector_type(8))) float v8f;

#define B_SZ    256
#define N_IN    512
#define NQ      12
#define NL      2
#define DIM     4096      // 2^12 amplitudes
#define QTHREADS 128
#define PER     (DIM / QTHREADS)   // 32 amplitudes per thread

// ---------------------------------------------------------------------------
// Kernel 1: h = relu(x @ W^T + b), f32 WMMA 16x16x4.
// Grid: 16 waves (one per 16-row tile of x). N padded 12 -> 16 with zeros.
//
// V_WMMA_F32_16X16X4_F32 operand layout (ISA 7.12.2):
//   A (16x4, 2 VGPRs): lane L: M = L%16; VGPR0 = A[M][2*(L/16)],
//                                         VGPR1 = A[M][2*(L/16)+1]
//   B (4x16, 2 VGPRs): lane L: N = L%16; VGPR0 = B[2*(L/16)][N],
//                                         VGPR1 = B[2*(L/16)+1][N]
//   C/D (16x16, 8 VGPRs): VGPR r, lane L: M = r + 8*(L/16), N = L%16
// ---------------------------------------------------------------------------
__global__ void __launch_bounds__(32)
gemm_h_wmma(const float* __restrict__ x,      // [256][512]
            const float* __restrict__ W,      // [12][512]
            const float* __restrict__ bias,   // [12]
            float* __restrict__ h)            // [256][16] (padded)
{
    const int tile = blockIdx.x;          // 0..15
    const int lane = threadIdx.x;         // 0..31
    const int m    = lane & 15;
    const int hi   = lane >> 4;           // 0 or 1 (K half-select)
    const int row  = tile * 16 + m;
    const int n    = lane & 15;           // B fragment column

    // Zero-mask padded columns without diverging (EXEC must be all-1s at WMMA).
    const int   nn   = (n < NQ) ? n : 0;
    const float mask = (n < NQ) ? 1.0f : 0.0f;

    v8f acc = {};
    for (int k = 0; k < N_IN; k += 4) {
        v2f a, bf;
        a.x  = x[row * N_IN + k + 2 * hi];
        a.y  = x[row * N_IN + k + 2 * hi + 1];
        bf.x = W[nn * N_IN + k + 2 * hi]     * mask;   // B[k'][n] = W[n][k']
        bf.y = W[nn * N_IN + k + 2 * hi + 1] * mask;
        acc = __builtin_amdgcn_wmma_f32_16x16x4_f32(
                  /*neg_a=*/false, a, /*neg_b=*/false, bf,
                  /*c_mod=*/(short)0, acc, /*reuse_a=*/false, /*reuse_b=*/false);
    }

    const float bv = (n < NQ) ? bias[n] : 0.0f;
#pragma unroll
    for (int r = 0; r < 8; ++r) {
        const int M = r + 8 * hi;
        if (n < NQ) {
            float v = acc[r] + bv;
            h[(tile * 16 + M) * 16 + n] = fmaxf(v, 0.0f);
        }
    }
}

// ---------------------------------------------------------------------------
// Kernel 2: 12-qubit statevector sim, one block per batch sample.
// State kept in LDS (re/im, 32 KB). Per layer per wire apply fused
// U = RZ(p1) * RY(h + p0); then the whole CNOT ladder as one Gray-code
// permutation state'[i] = state[i ^ (i>>1)].
// ---------------------------------------------------------------------------
__global__ void __launch_bounds__(QTHREADS)
qsim12(const float* __restrict__ h,       // [256][16]
       const float* __restrict__ param,   // [12][2][2]
       float* __restrict__ out)           // [256][12]
{
    __shared__ float re[DIM];
    __shared__ float im[DIM];

    const int b   = blockIdx.x;
    const int tid = threadIdx.x;

    // init |0...0>
    for (int i = tid; i < DIM; i += QTHREADS) { re[i] = 0.0f; im[i] = 0.0f; }
    __syncthreads();
    if (tid == 0) re[0] = 1.0f;
    __syncthreads();

    for (int l = 0; l < NL; ++l) {
        for (int j = 0; j < NQ; ++j) {
            const float theta = h[b * 16 + j] + param[(j * NL + l) * 2 + 0];
            const float phi   = param[(j * NL + l) * 2 + 1];
            float sy, cy, sz, cz;
            sincosf(0.5f * theta, &sy, &cy);
            sincosf(0.5f * phi,   &sz, &cz);
            // U = RZ(phi) * RY(theta)
            const float u00r =  cy * cz, u00i = -cy * sz;
            const float u01r = -sy * cz, u01i =  sy * sz;
            const float u10r =  sy * cz, u10i =  sy * sz;
            const float u11r =  cy * cz, u11i =  cy * sz;

            const int sh     = 11 - j;
            const int stride = 1 << sh;
#pragma unroll 4
            for (int p = tid; p < DIM / 2; p += QTHREADS) {
                const int i0 = ((p >> sh) << (sh + 1)) | (p & (stride - 1));
                const int i1 = i0 | stride;
                const float a0r = re[i0], a0i = im[i0];
                const float a1r = re[i1], a1i = im[i1];
                re[i0] = u00r * a0r - u00i * a0i + u01r * a1r - u01i * a1i;
                im[i0] = u00r * a0i + u00i * a0r + u01r * a1i + u01i * a1r;
                re[i1] = u10r * a0r - u10i * a0i + u11r * a1r - u11i * a1i;
                im[i1] = u10r * a0i + u10i * a0r + u11r * a1i + u11i * a1r;
            }
            __syncthreads();
        }

        // CNOT ladder j->j+1 for j=0..10  ==  state'[idx] = state[idx ^ (idx>>1)]
        float tr[PER], ti[PER];
#pragma unroll
        for (int k = 0; k < PER; ++k) {
            const int idx = tid * PER + k;
            const int src = idx ^ (idx >> 1);
            tr[k] = re[src];
            ti[k] = im[src];
        }
        __syncthreads();
#pragma unroll
        for (int k = 0; k < PER; ++k) {
            const int idx = tid * PER + k;
            re[idx] = tr[k];
            im[idx] = ti[k];
        }
        __syncthreads();
    }

    // Measurement: out[b][i] = sum_{idx: bit(11-i)==0} |amp|^2
    float acc[NQ];
#pragma unroll
    for (int i = 0; i < NQ; ++i) acc[i] = 0.0f;
    for (int k = 0; k < PER; ++k) {
        const int idx = tid * PER + k;
        const float p = re[idx] * re[idx] + im[idx] * im[idx];
#pragma unroll
        for (int i = 0; i < NQ; ++i)
            if (((idx >> (11 - i)) & 1) == 0) acc[i] += p;
    }
    __syncthreads();   // state no longer needed; reuse re[] as reduce scratch

#pragma unroll
    for (int i = 0; i < NQ; ++i) re[i * QTHREADS + tid] = acc[i];
    __syncthreads();
    for (int off = QTHREADS / 2; off > 0; off >>= 1) {
        if (tid < off) {
#pragma unroll
            for (int i = 0; i < NQ; ++i)
                re[i * QTHREADS + tid] += re[i * QTHREADS + tid + off];
        }
        __syncthreads();
    }
    if (tid < NQ) out[b * NQ + tid] = re[tid * QTHREADS];
}

// ---------------------------------------------------------------------------
extern "C" void kernel_launch(void* const* d_in, const int* in_sizes, int n_in,
                              void* d_out, int out_size, void* d_ws, size_t ws_size,
                              hipStream_t stream)
{
    (void)in_sizes; (void)n_in; (void)out_size; (void)ws_size;
    const float* x     = (const float*)d_in[0];   // (256,512)
    const float* W     = (const float*)d_in[1];   // (12,512)
    const float* bias  = (const float*)d_in[2];   // (12,)
    const float* param = (const float*)d_in[3];   // (12,2,2)
    float*       out   = (float*)d_out;           // (256,12)
    float*       hbuf  = (float*)d_ws;            // (256,16) padded

    gemm_h_wmma<<<B_SZ / 16, 32, 0, stream>>>(x, W, bias, hbuf);
    qsim12<<<B_SZ, QTHREADS, 0, stream>>>(hbuf, param, out);
}